// test_65644280152995
// MI455X (gfx1250) — compile-verified
//
#include <hip/hip_runtime.h>
#include <math.h>

// ---------------- problem constants ----------------
#define EPSBN 1e-5f
#define SLOPE 0.01f
static constexpr int  HP  = 24;                       // half-res per dim
static constexpr long P   = 2L*24*24*24*24;           // 663552 output positions of conv2
static constexpr long NX  = 2L*48*48*48*48;           // 10,616,832 x elements
static constexpr int  RX_BLOCKS = 648;                // 648*256*16*4 == NX
static constexpr int  M3_BLOCKS = 216;                // 216*256*12 == P
static constexpr int  POS_BLOCKS = 2592;              // 2592*256 == P

// workspace byte offsets
static constexpr size_t OFF_U2   = 0;                        // P*8 floats (padded [pos][8])
static constexpr size_t OFF_WP   = (size_t)P*8*4;            // 21,233,664 : 768 floats (B frags)
static constexpr size_t OFF_DACC = OFF_WP + 768*4;           // 21,236,736 : 56 doubles
static constexpr size_t OFF_FC   = OFF_DACC + 512;           // 24 floats (affine consts)

typedef float vf2 __attribute__((ext_vector_type(2)));
typedef float vf8 __attribute__((ext_vector_type(8)));

__device__ __forceinline__ float lrelu(float v){ return v >= 0.f ? v : SLOPE*v; }

// ---------------- 1) prep: zero accumulators + build WMMA B fragments ----------------
// W[K=48][N=16]: K = c*16 + (p*8+q*4+r*2+s), N = out channel (cols 6..15 zero).
// B fragment layout (mirror of documented 32-bit A 16x4 layout):
//   lane = n + 16*h ; chunk t: vgpr0 = W[4t+2h][n], vgpr1 = W[4t+2h+1][n]
__global__ void k_prep(const float* __restrict__ w2, float* __restrict__ wp,
                       double* __restrict__ dacc)
{
    int tid = threadIdx.x;
    if (tid < 56) dacc[tid] = 0.0;
    if (tid < 384) {
        int n = tid & 15, h = (tid >> 4) & 1, t = tid >> 5;
        int K0 = 4*t + 2*h;                          // K and K+1 go into (x,y)
        float vx = (n < 6) ? w2[n*48 + K0]     : 0.f;
        float vy = (n < 6) ? w2[n*48 + K0 + 1] : 0.f;
        ((float2*)wp)[t*32 + h*16 + n] = make_float2(vx, vy);
    }
}

// ---------------- 2) global sum / sumsq of x ----------------
__global__ __launch_bounds__(256) void k_rx(const float* __restrict__ x,
                                            double* __restrict__ dacc)
{
    __shared__ double sacc[2];
    int tid = threadIdx.x;
    if (tid < 2) sacc[tid] = 0.0;
    __syncthreads();
    long gid = (long)blockIdx.x*256 + tid;
    float s = 0.f, q = 0.f;
    #pragma unroll
    for (int it = 0; it < 16; ++it) {
        long idx = gid + (long)it * (RX_BLOCKS*256);
        float4 v = *(const float4*)(x + idx*4);
        s += v.x + v.y + v.z + v.w;
        q += v.x*v.x + v.y*v.y + v.z*v.z + v.w*v.w;
    }
    atomicAdd(&sacc[0], (double)s);
    atomicAdd(&sacc[1], (double)q);
    __syncthreads();
    if (tid < 2) atomicAdd(&dacc[tid], sacc[tid]);
}

// ---------------- 3) BN1 affine constants ----------------
__global__ void k_fin1(const double* __restrict__ dacc, const float* __restrict__ w1,
                       const float* __restrict__ g1, const float* __restrict__ b1,
                       float* __restrict__ fc)
{
    if (threadIdx.x != 0) return;
    double n  = (double)NX;
    double mu = dacc[0]/n;
    double var = dacc[1]/n - mu*mu;
    for (int c = 0; c < 3; ++c) {
        double wc  = (double)w1[c];
        double inv = 1.0 / sqrt(wc*wc*var + (double)EPSBN);
        fc[c]   = (float)(wc*inv) * g1[c];                       // a_c
        fc[3+c] = (float)(-wc*mu*inv) * g1[c] + b1[c];           // c_c
    }
}

// ---------------- 4) conv2 via V_WMMA_F32_16X16X4_F32 ----------------
// Each block: 256 positions. LDS holds per-position 16 x-values (stride 18 -> no bank conflicts).
// Per wave: 2 tiles of 16 positions; 12 chained K=4 WMMAs (K=48 = 3 ch * 16 taps).
__global__ __launch_bounds__(256) void k_conv2(const float* __restrict__ x,
                                               const float* __restrict__ wp,
                                               const float* __restrict__ fc,
                                               float* __restrict__ u2,
                                               double* __restrict__ dacc)
{
    __shared__ float lx[256*18];
    __shared__ float lsum[6], lssq[6];
    int tid = threadIdx.x;
    if (tid < 6) { lsum[tid] = 0.f; lssq[tid] = 0.f; }

    int pos = blockIdx.x*256 + tid;
    int l  = pos % HP;  int t1 = pos / HP;
    int kk = t1 % HP;   t1 /= HP;
    int j  = t1 % HP;   t1 /= HP;
    int i  = t1 % HP;   int b = t1 / HP;
    long xbase = ((((long)b*48 + 2*i)*48 + 2*j)*48 + 2*kk)*48 + 2*l;

    float* row = &lx[tid*18];
    #pragma unroll
    for (int p = 0; p < 2; ++p)
      #pragma unroll
      for (int q = 0; q < 2; ++q)
        #pragma unroll
        for (int r = 0; r < 2; ++r) {
            float2 v = *(const float2*)(x + xbase + (long)p*110592 + q*2304 + r*48);
            *(float2*)(row + (p*8 + q*4 + r*2)) = v;
        }

    int lane = tid & 31, l16 = lane & 15, h = lane >> 4, wv = tid >> 5;

    vf2 Bfrag[12];
    #pragma unroll
    for (int t = 0; t < 12; ++t)
        Bfrag[t] = *(const vf2*)(wp + ((t*2 + h)*16 + l16)*2);

    float a1[3], c1[3];
    #pragma unroll
    for (int c = 0; c < 3; ++c) { a1[c] = fc[c]; c1[c] = fc[3+c]; }

    __syncthreads();

    #pragma unroll
    for (int tau = 0; tau < 2; ++tau) {
        int bl = wv*32 + tau*16;                 // tile base (local position index)
        vf8 acc = {0.f,0.f,0.f,0.f,0.f,0.f,0.f,0.f};
        #pragma unroll
        for (int t = 0; t < 12; ++t) {
            int c  = t >> 2;
            int fb = (t & 3)*4 + 2*h;
            float2 xv = *(const float2*)&lx[(bl + l16)*18 + fb];
            vf2 A;
            A.x = lrelu(a1[c]*xv.x + c1[c]);
            A.y = lrelu(a1[c]*xv.y + c1[c]);
            acc = __builtin_amdgcn_wmma_f32_16x16x4_f32(
                      false, A, false, Bfrag[t], (short)0, acc, false, false);
        }
        // D layout: vgpr r, lane<16 -> (M=r, N=lane); lane>=16 -> (M=r+8, N=lane-16)
        if (l16 < 6) {
            long pbase = (long)blockIdx.x*256 + bl + 8*h;
            float ls = 0.f, lq = 0.f;
            #pragma unroll
            for (int r = 0; r < 8; ++r) {
                float v = acc[r];
                u2[(pbase + r)*8 + l16] = v;
                ls += v; lq += v*v;
            }
            atomicAdd(&lsum[l16], ls);
            atomicAdd(&lssq[l16], lq);
        }
    }
    __syncthreads();
    if (tid < 6) {
        atomicAdd(&dacc[2 + tid], (double)lsum[tid]);
        atomicAdd(&dacc[8 + tid], (double)lssq[tid]);
    }
}

// ---------------- 5) BN2 affine constants ----------------
__global__ void k_fin2(const double* __restrict__ dacc, const float* __restrict__ g2,
                       const float* __restrict__ b2, float* __restrict__ fc)
{
    if (threadIdx.x != 0) return;
    double n = (double)P;
    for (int c = 0; c < 6; ++c) {
        double mu  = dacc[2+c]/n;
        double var = dacc[8+c]/n - mu*mu;
        double inv = 1.0 / sqrt(var + (double)EPSBN);
        float sc = (float)inv * g2[c];
        fc[6+c]  = sc;
        fc[12+c] = b2[c] - (float)mu*sc;
    }
}

// ---------------- 6) moments of y2 = lrelu(bn2(u2)): mean[6], upper-tri S[21] ----------------
__global__ __launch_bounds__(256) void k_m3(const float* __restrict__ u2,
                                            const float* __restrict__ fc,
                                            double* __restrict__ dacc)
{
    __shared__ double sacc[27];
    int tid = threadIdx.x;
    if (tid < 27) sacc[tid] = 0.0;
    __syncthreads();
    float m[6]; float s[21];
    #pragma unroll
    for (int c = 0; c < 6; ++c) m[c] = 0.f;
    #pragma unroll
    for (int k = 0; k < 21; ++k) s[k] = 0.f;
    long gid = (long)blockIdx.x*256 + tid;
    for (int it = 0; it < 12; ++it) {
        long pos = gid + (long)it*(M3_BLOCKS*256);
        float4 ua = *(const float4*)(u2 + pos*8);
        float4 ub = *(const float4*)(u2 + pos*8 + 4);
        float uv[6] = {ua.x, ua.y, ua.z, ua.w, ub.x, ub.y};
        float y[6];
        #pragma unroll
        for (int c = 0; c < 6; ++c) y[c] = lrelu(uv[c]*fc[6+c] + fc[12+c]);
        int idx = 0;
        #pragma unroll
        for (int c = 0; c < 6; ++c) {
            m[c] += y[c];
            #pragma unroll
            for (int c2 = c; c2 < 6; ++c2) s[idx++] += y[c]*y[c2];
        }
    }
    #pragma unroll
    for (int c = 0; c < 6; ++c)  atomicAdd(&sacc[c],     (double)m[c]);
    #pragma unroll
    for (int k = 0; k < 21; ++k) atomicAdd(&sacc[6 + k], (double)s[k]);
    __syncthreads();
    if (tid < 27) atomicAdd(&dacc[14 + tid], sacc[tid]);
}

// ---------------- 7) analytic BN3 constants ----------------
__global__ void k_fin3(const double* __restrict__ dacc, const float* __restrict__ wt,
                       const float* __restrict__ g3, const float* __restrict__ b3,
                       float* __restrict__ fc)
{
    if (threadIdx.x != 0) return;
    double n = (double)P;
    double mean[6], S[6][6];
    for (int c = 0; c < 6; ++c) mean[c] = dacc[14+c]/n;
    int idx = 0;
    for (int c = 0; c < 6; ++c)
        for (int c2 = c; c2 < 6; ++c2) {
            double v = dacc[20+idx]/n; S[c][c2] = v; S[c2][c] = v; ++idx;
        }
    for (int o = 0; o < 3; ++o) {
        double mu = 0.0, m2 = 0.0;
        for (int f = 0; f < 16; ++f) {
            double wv[6];
            for (int c = 0; c < 6; ++c) wv[c] = (double)wt[(c*3+o)*16 + f];
            for (int c = 0; c < 6; ++c) {
                mu += wv[c]*mean[c];
                for (int c2 = 0; c2 < 6; ++c2) m2 += wv[c]*wv[c2]*S[c][c2];
            }
        }
        mu /= 16.0; m2 /= 16.0;
        double var = m2 - mu*mu;
        double inv = 1.0 / sqrt(var + (double)EPSBN);
        float A = (float)inv * g3[o];
        fc[18+o] = A;
        fc[21+o] = b3[o] - (float)mu*A;
    }
}

// ---------------- 8) transpose-conv expansion + BN3 + sigmoid (bandwidth kernel) ----------------
__global__ __launch_bounds__(256) void k_out(const float* __restrict__ u2,
                                             const float* __restrict__ wt,
                                             const float* __restrict__ fc,
                                             float* __restrict__ out)
{
    int pos = blockIdx.x*256 + threadIdx.x;
    int l  = pos % HP;  int t1 = pos / HP;
    int kk = t1 % HP;   t1 /= HP;
    int j  = t1 % HP;   t1 /= HP;
    int i  = t1 % HP;   int b = t1 / HP;

    float4 ua = *(const float4*)(u2 + (long)pos*8);
    float4 ub = *(const float4*)(u2 + (long)pos*8 + 4);
    float uv[6] = {ua.x, ua.y, ua.z, ua.w, ub.x, ub.y};
    float y[6];
    #pragma unroll
    for (int c = 0; c < 6; ++c) y[c] = lrelu(uv[c]*fc[6+c] + fc[12+c]);

    float acc[48];
    #pragma unroll
    for (int u = 0; u < 48; ++u) acc[u] = 0.f;
    #pragma unroll
    for (int c = 0; c < 6; ++c)
      #pragma unroll
      for (int o = 0; o < 3; ++o) {
        const float* wrow = wt + (c*3 + o)*16;   // uniform address -> SMEM s_loads
        #pragma unroll
        for (int f = 0; f < 16; ++f) acc[o*16 + f] += y[c]*wrow[f];
      }

    #pragma unroll
    for (int o = 0; o < 3; ++o) {
        float A3 = fc[18+o], B3 = fc[21+o];
        #pragma unroll
        for (int p = 0; p < 2; ++p)
          #pragma unroll
          for (int q = 0; q < 2; ++q)
            #pragma unroll
            for (int r = 0; r < 2; ++r) {
                int f = p*8 + q*4 + r*2;
                float z0 = acc[o*16 + f    ]*A3 + B3;
                float z1 = acc[o*16 + f + 1]*A3 + B3;
                float s0 = 1.f / (1.f + __expf(-z0));
                float s1 = 1.f / (1.f + __expf(-z1));
                long oidx = (((((long)b*3 + o)*48 + (2*i+p))*48 + (2*j+q))*48 + (2*kk+r))*48 + 2*l;
                *(float2*)(out + oidx) = make_float2(s0, s1);
            }
    }
}

// ---------------- launch ----------------
extern "C" void kernel_launch(void* const* d_in, const int* in_sizes, int n_in,
                              void* d_out, int out_size, void* d_ws, size_t ws_size,
                              hipStream_t stream)
{
    (void)in_sizes; (void)n_in; (void)out_size; (void)ws_size;
    const float* x  = (const float*)d_in[0];
    const float* w1 = (const float*)d_in[1];
    const float* g1 = (const float*)d_in[2];
    const float* b1 = (const float*)d_in[3];
    const float* w2 = (const float*)d_in[4];
    const float* g2 = (const float*)d_in[5];
    const float* b2 = (const float*)d_in[6];
    const float* wt = (const float*)d_in[7];
    const float* g3 = (const float*)d_in[8];
    const float* b3 = (const float*)d_in[9];
    float*  out  = (float*)d_out;
    float*  u2   = (float*)((char*)d_ws + OFF_U2);
    float*  wp   = (float*)((char*)d_ws + OFF_WP);
    double* dacc = (double*)((char*)d_ws + OFF_DACC);
    float*  fc   = (float*)((char*)d_ws + OFF_FC);

    k_prep <<<1, 512, 0, stream>>>(w2, wp, dacc);
    k_rx   <<<RX_BLOCKS, 256, 0, stream>>>(x, dacc);
    k_fin1 <<<1, 32, 0, stream>>>(dacc, w1, g1, b1, fc);
    k_conv2<<<POS_BLOCKS, 256, 0, stream>>>(x, wp, fc, u2, dacc);
    k_fin2 <<<1, 32, 0, stream>>>(dacc, g2, b2, fc);
    k_m3   <<<M3_BLOCKS, 256, 0, stream>>>(u2, fc, dacc);
    k_fin3 <<<1, 32, 0, stream>>>(dacc, wt, g3, b3, fc);
    k_out  <<<POS_BLOCKS, 256, 0, stream>>>(u2, wt, fc, out);
}